// MultiHeadAttention_30236569764221
// MI455X (gfx1250) — compile-verified
//
#include <hip/hip_runtime.h>

#define B_SZ 2
#define N_SEQ 384
#define DIM 256
#define EDIM 128
#define HEADS 8
#define DH 64
#define INNER 512
#define ROWS (B_SZ * N_SEQ)          // 768
#define TI 16
#define TJ 16
#define CPAD 132                     // 128 + 4 (16B aligned, bank-spread)
#define DPAD 68                      // 64 + 4

typedef float v8f __attribute__((ext_vector_type(8)));
typedef float v2f __attribute__((ext_vector_type(2)));
typedef int v4i_vs __attribute__((vector_size(16)));

#if defined(__has_builtin)
#if __has_builtin(__builtin_amdgcn_global_load_async_to_lds_b128) && \
    __has_builtin(__builtin_amdgcn_s_wait_asynccnt)
#define USE_ASYNC_LDS 1
#endif
#endif

#ifdef USE_ASYNC_LDS
#define GLOBAL_AS __attribute__((address_space(1)))
#define LDS_AS __attribute__((address_space(3)))
static __device__ __forceinline__ void async_cp16(const float* g, float* l) {
  __builtin_amdgcn_global_load_async_to_lds_b128((GLOBAL_AS v4i_vs*)g,
                                                 (LDS_AS v4i_vs*)l, 0, 0);
}
#endif

static __device__ __forceinline__ v8f wmma4(v2f a, v2f b, v8f c) {
  // V_WMMA_F32_16X16X4_F32 : D = A(16x4) x B(4x16) + C(16x16)
  return __builtin_amdgcn_wmma_f32_16x16x4_f32(false, a, false, b, (short)0, c,
                                               false, false);
}

// ---------------------------------------------------------------------------
// K1: qkv = nodes @ [Wq | Wkv] + [bq | bkv]   (M=768, N=1536, K=256)
// one wave per 16x16 tile; 48*96 = 4608 tiles / 8 waves = 576 blocks
// ---------------------------------------------------------------------------
__global__ __launch_bounds__(256) void proj_qkv_kernel(
    const float* __restrict__ nodes, const float* __restrict__ Wq,
    const float* __restrict__ bq, const float* __restrict__ Wkv,
    const float* __restrict__ bkv, float* __restrict__ qkv) {
  const int tile = blockIdx.x * 8 + (threadIdx.x >> 5);
  const int lane = threadIdx.x & 31;
  const int mt = tile / 96, nt = tile % 96;
  const int m0 = mt * 16, n0 = nt * 16;
  const int mrow = lane & 15;
  const int koff = (lane < 16) ? 0 : 2;

  const float* W;
  int ldb, ncol;
  if (n0 < 512) { W = Wq;  ldb = 512;  ncol = n0; }
  else          { W = Wkv; ldb = 1024; ncol = n0 - 512; }

  v8f acc = {0.f, 0.f, 0.f, 0.f, 0.f, 0.f, 0.f, 0.f};
  const float* arow = &nodes[(long)(m0 + mrow) * DIM];
#pragma unroll 4
  for (int kk = 0; kk < DIM; kk += 4) {
    float2 av = *(const float2*)&arow[kk + koff];
    v2f a; a.x = av.x; a.y = av.y;
    v2f b;
    b.x = W[(long)(kk + koff) * ldb + ncol + mrow];
    b.y = W[(long)(kk + koff + 1) * ldb + ncol + mrow];
    acc = wmma4(a, b, acc);
  }
  const float bias = (n0 < 512) ? bq[ncol + mrow] : bkv[ncol + mrow];
#pragma unroll
  for (int r = 0; r < 8; ++r) {
    int mr = r + ((lane < 16) ? 0 : 8);
    qkv[(long)(m0 + mr) * 1536 + n0 + mrow] = acc[r] + bias;
  }
}

// ---------------------------------------------------------------------------
// K2: RoPE on q (cols 0..511, plus scale d^-1/2) and k (cols 512..1023)
// grid 768 blocks x 256 threads (thread = head*32 + pair)
// ---------------------------------------------------------------------------
__global__ __launch_bounds__(256) void rope_kernel(float* __restrict__ qkv) {
  const int row = blockIdx.x;              // 0..767
  const int t = threadIdx.x;
  const int h = t >> 5, m = t & 31;        // pair m -> dims 2m, 2m+1
  const int pos = row % N_SEQ;
  const float inv = __powf(10000.0f, -(float)(2 * m) / 64.0f);
  float s, c;
  __sincosf((float)pos * inv, &s, &c);
  const float scale = 0.125f;              // 64^-0.5
  float* q = &qkv[(long)row * 1536 + h * DH + 2 * m];
  float* k = q + 512;
  float q0 = q[0], q1 = q[1], k0 = k[0], k1 = k[1];
  q[0] = (q0 * c - q1 * s) * scale;
  q[1] = (q1 * c + q0 * s) * scale;
  k[0] = k0 * c - k1 * s;
  k[1] = k1 * c + k0 * s;
}

// ---------------------------------------------------------------------------
// K3: w[(b*8+h)*384+i][c] = sum_d q_scaled[b,i,h,d] * We[c, h*64+d]
// M = 6144 (384 tiles), N = 128 (8 tiles), K = 64 -> 3072 wave jobs -> 384 blk
// ---------------------------------------------------------------------------
__global__ __launch_bounds__(256) void proj_w_kernel(
    const float* __restrict__ qkv, const float* __restrict__ We,
    float* __restrict__ wbuf) {
  const int tile = blockIdx.x * 8 + (threadIdx.x >> 5);
  const int lane = threadIdx.x & 31;
  const int mt = tile >> 3, nt = tile & 7;
  const int bhrow = mt * 16;               // 0..6143
  const int bh = bhrow / N_SEQ;
  const int b = bh >> 3, h = bh & 7;
  const int i0 = bhrow % N_SEQ;
  const int c0 = nt * 16;
  const int mrow = lane & 15;
  const int koff = (lane < 16) ? 0 : 2;

  v8f acc = {0.f, 0.f, 0.f, 0.f, 0.f, 0.f, 0.f, 0.f};
  const float* arow = &qkv[(long)(b * N_SEQ + i0 + mrow) * 1536 + h * DH];
  const float* brow = &We[(long)(c0 + mrow) * INNER + h * DH];
#pragma unroll
  for (int kk = 0; kk < DH; kk += 4) {
    float2 av = *(const float2*)&arow[kk + koff];
    float2 bv = *(const float2*)&brow[kk + koff];
    v2f a; a.x = av.x; a.y = av.y;
    v2f bb; bb.x = bv.x; bb.y = bv.y;
    acc = wmma4(a, bb, acc);
  }
#pragma unroll
  for (int r = 0; r < 8; ++r) {
    int mr = r + ((lane < 16) ? 0 : 8);
    wbuf[(long)(bhrow + mr) * EDIM + c0 + mrow] = acc[r];
  }
}

// K3b: qbe[(b*8+h)*384+i] = sum_d q_scaled * be[h*64+d]
__global__ void qbe_kernel(const float* __restrict__ qkv,
                           const float* __restrict__ be,
                           float* __restrict__ qbe) {
  int idx = blockIdx.x * 256 + threadIdx.x;
  if (idx >= B_SZ * HEADS * N_SEQ) return;
  int bh = idx / N_SEQ, b = bh >> 3, h = bh & 7, i = idx % N_SEQ;
  const float* q = &qkv[(long)(b * N_SEQ + i) * 1536 + h * DH];
  const float* bb = &be[h * DH];
  float s = 0.f;
#pragma unroll
  for (int d = 0; d < DH; ++d) s += q[d] * bb[d];
  qbe[idx] = s;
}

// ---------------------------------------------------------------------------
// K4: fused edge-biased flash attention. One WG per (b, h, 16-query tile).
//   s[i,j] = q_i.k_j + edges[i,j].w_i + q_i.be      (VALU, per-i GEMV shape)
//   online softmax, g[i,c] += p~ * edges[i,j,c]     (VALU, registers)
//   o += p~ @ v                                     (WMMA, 4 ntiles x 2 khalf)
//   out = (o + g @ We)/l + be                       (WMMA epilogue)
// Edges/k/v tiles staged with GLOBAL_LOAD_ASYNC_TO_LDS_B128 when available.
// ---------------------------------------------------------------------------
__global__ __launch_bounds__(256) void attn_kernel(
    const float* __restrict__ edges, const float* __restrict__ qkv,
    const float* __restrict__ wbuf, const float* __restrict__ qbe,
    const float* __restrict__ We, const float* __restrict__ be,
    float* __restrict__ ao) {
  __shared__ float eds[TI * TJ * CPAD];    // 135168 B
  __shared__ float q_l[TI * DPAD];
  __shared__ float k_l[TJ * DPAD];
  __shared__ float v_l[TJ * DPAD];
  __shared__ float w_l[TI * CPAD];
  __shared__ float p_l[TI * 17];
  __shared__ float m_l[TI], l_l[TI], al_l[TI], qb_l[TI];

  const int it = blockIdx.x, h = blockIdx.y, b = blockIdx.z;
  const int tid = threadIdx.x;
  const int lane = tid & 31;
  const int wv = tid >> 5;
  const int i0 = it * TI;
  const int row0 = b * N_SEQ + i0;
  const int bh0 = (b * HEADS + h) * N_SEQ + i0;

  { // stage q (roped+scaled)
    int i = tid >> 4, d = (tid & 15) * 4;
    *(float4*)&q_l[i * DPAD + d] =
        *(const float4*)&qkv[(long)(row0 + i) * 1536 + h * DH + d];
  }
  { // stage w
    int i = tid >> 4, c = (tid & 15) * 8;
    const float* src = &wbuf[(long)(bh0 + i) * EDIM + c];
    *(float4*)&w_l[i * CPAD + c] = *(const float4*)&src[0];
    *(float4*)&w_l[i * CPAD + c + 4] = *(const float4*)&src[4];
  }
  if (tid < TI) {
    qb_l[tid] = qbe[bh0 + tid];
    m_l[tid] = -3.0e38f;
    l_l[tid] = 0.0f;
  }
  v8f oacc = {0.f, 0.f, 0.f, 0.f, 0.f, 0.f, 0.f, 0.f};
  float4 g0 = {0.f, 0.f, 0.f, 0.f}, g1 = {0.f, 0.f, 0.f, 0.f};
  __syncthreads();

  for (int j0 = 0; j0 < N_SEQ; j0 += TJ) {
    { // A: edges tile -> LDS (thread = (i,j), 128 contiguous floats each)
      int i = tid >> 4, j = tid & 15;
      const float* src =
          &edges[(((long)(b * N_SEQ + i0 + i)) * N_SEQ + (j0 + j)) * EDIM];
      float* dst = &eds[(i * TJ + j) * CPAD];
#pragma unroll
      for (int c = 0; c < EDIM; c += 4) {
#ifdef USE_ASYNC_LDS
        async_cp16(&src[c], &dst[c]);
#else
        *(float4*)&dst[c] = *(const float4*)&src[c];
#endif
      }
    }
    { // A: k,v tiles
      int t = tid & 127;
      int j = t >> 3, d = (t & 7) * 8;
      int col = ((tid < 128) ? 512 : 1024) + h * DH + d;
      const float* srow = &qkv[(long)(b * N_SEQ + j0 + j) * 1536 + col];
      float* drow = ((tid < 128) ? k_l : v_l) + j * DPAD + d;
#ifdef USE_ASYNC_LDS
      async_cp16(&srow[0], &drow[0]);
      async_cp16(&srow[4], &drow[4]);
#else
      *(float4*)&drow[0] = *(const float4*)&srow[0];
      *(float4*)&drow[4] = *(const float4*)&srow[4];
#endif
    }
#ifdef USE_ASYNC_LDS
    __builtin_amdgcn_s_wait_asynccnt(0);
#endif
    __syncthreads();

    { // B: scores
      int i = tid >> 4, j = tid & 15;
      float acc = qb_l[i];
      const float* qq = &q_l[i * DPAD];
      const float* kk = &k_l[j * DPAD];
#pragma unroll
      for (int d = 0; d < DH; d += 4) {
        float4 a = *(const float4*)&qq[d];
        float4 bb = *(const float4*)&kk[d];
        acc += a.x * bb.x + a.y * bb.y + a.z * bb.z + a.w * bb.w;
      }
      const float* ww = &w_l[i * CPAD];
      const float* ee = &eds[(i * TJ + j) * CPAD];
#pragma unroll
      for (int c = 0; c < EDIM; c += 4) {
        float4 a = *(const float4*)&ww[c];
        float4 bb = *(const float4*)&ee[c];
        acc += a.x * bb.x + a.y * bb.y + a.z * bb.z + a.w * bb.w;
      }
      p_l[i * 17 + j] = acc;
    }
    __syncthreads();

    // C1: row max + alpha
    if (tid < TI) {
      float mo = m_l[tid], mt = mo;
#pragma unroll
      for (int j = 0; j < TJ; ++j) mt = fmaxf(mt, p_l[tid * 17 + j]);
      m_l[tid] = mt;
      al_l[tid] = __expf(mo - mt);
    }
    __syncthreads();

    // C2: exponentiate + rescale partial accumulators
    {
      int i = tid >> 4, j = tid & 15;
      float a = al_l[i];
      p_l[i * 17 + j] = __expf(p_l[i * 17 + j] - m_l[i]);
      g0.x *= a; g0.y *= a; g0.z *= a; g0.w *= a;
      g1.x *= a; g1.y *= a; g1.z *= a; g1.w *= a;
    }
#pragma unroll
    for (int r = 0; r < 8; ++r) {
      float a = (lane < 16) ? al_l[r] : al_l[r + 8];
      oacc[r] *= a;
    }
    __syncthreads();

    // C3: running denominator
    if (tid < TI) {
      float s = 0.f;
#pragma unroll
      for (int j = 0; j < TJ; ++j) s += p_l[tid * 17 + j];
      l_l[tid] = l_l[tid] * al_l[tid] + s;
    }
    // D1: g += p~ * edges  (thread = (i, c-octet))
    {
      int i = tid >> 4;
      const float* ee = &eds[i * TJ * CPAD + (tid & 15) * 8];
      const float* pp = &p_l[i * 17];
#pragma unroll
      for (int j = 0; j < TJ; ++j) {
        float p = pp[j];
        float4 e0 = *(const float4*)&ee[j * CPAD];
        float4 e1 = *(const float4*)&ee[j * CPAD + 4];
        g0.x += p * e0.x; g0.y += p * e0.y; g0.z += p * e0.z; g0.w += p * e0.w;
        g1.x += p * e1.x; g1.y += p * e1.y; g1.z += p * e1.z; g1.w += p * e1.w;
      }
    }
    // D2: o += p~ @ v  (WMMA; wave -> (ntile, khalf))
    {
      int nt = wv & 3, kh = wv >> 2;
      int mrow = lane & 15;
      int koff = (lane < 16) ? 0 : 2;
#pragma unroll
      for (int s = 2 * kh; s < 2 * kh + 2; ++s) {
        v2f a, bb;
        a.x = p_l[mrow * 17 + s * 4 + koff];
        a.y = p_l[mrow * 17 + s * 4 + koff + 1];
        bb.x = v_l[(s * 4 + koff) * DPAD + nt * 16 + mrow];
        bb.y = v_l[(s * 4 + koff + 1) * DPAD + nt * 16 + mrow];
        oacc = wmma4(a, bb, oacc);
      }
    }
    __syncthreads();
  }

  // Finalize: alias post-loop buffers onto the (now dead) edge tile.
  float* g_l = eds;               // 16 x CPAD
  float* o_l = eds + TI * CPAD;   // 16 x DPAD
  {
    int i = tid >> 4, c = (tid & 15) * 8;
    *(float4*)&g_l[i * CPAD + c] = g0;
    *(float4*)&g_l[i * CPAD + c + 4] = g1;
  }
  if (wv < 4) {  // khalf 0 waves publish their partial o
    int nt = wv & 3;
#pragma unroll
    for (int r = 0; r < 8; ++r) {
      int mr = r + ((lane < 16) ? 0 : 8);
      o_l[mr * DPAD + nt * 16 + (lane & 15)] = oacc[r];
    }
  }
  __syncthreads();
  if (wv >= 4) {  // khalf 1 waves merge, add g@We, epilogue
    int nt = wv & 3;
    int mrow = lane & 15;
    int koff = (lane < 16) ? 0 : 2;
#pragma unroll
    for (int r = 0; r < 8; ++r) {
      int mr = r + ((lane < 16) ? 0 : 8);
      oacc[r] += o_l[mr * DPAD + nt * 16 + mrow];
    }
    const float* bcol = &We[(long)0 * INNER + h * DH + nt * 16 + mrow];
#pragma unroll 8
    for (int kk = 0; kk < EDIM; kk += 4) {
      v2f a, bb;
      a.x = g_l[mrow * CPAD + kk + koff];
      a.y = g_l[mrow * CPAD + kk + koff + 1];
      bb.x = bcol[(long)(kk + koff) * INNER];
      bb.y = bcol[(long)(kk + koff + 1) * INNER];
      oacc = wmma4(a, bb, oacc);
    }
    float bias = be[h * DH + nt * 16 + mrow];
#pragma unroll
    for (int r = 0; r < 8; ++r) {
      int mr = r + ((lane < 16) ? 0 : 8);
      float invl = 1.0f / l_l[mr];
      ao[(long)(row0 + mr) * INNER + h * DH + nt * 16 + mrow] =
          oacc[r] * invl + bias;
    }
  }
}

// ---------------------------------------------------------------------------
// K5: out = ao @ Wo + bo   (M=768, N=256, K=512) -> 768 wave jobs -> 96 blocks
// ---------------------------------------------------------------------------
__global__ __launch_bounds__(256) void out_proj_kernel(
    const float* __restrict__ ao, const float* __restrict__ Wo,
    const float* __restrict__ bo, float* __restrict__ out) {
  const int tile = blockIdx.x * 8 + (threadIdx.x >> 5);
  const int lane = threadIdx.x & 31;
  const int mt = tile >> 4, nt = tile & 15;
  const int m0 = mt * 16, n0 = nt * 16;
  const int mrow = lane & 15;
  const int koff = (lane < 16) ? 0 : 2;
  v8f acc = {0.f, 0.f, 0.f, 0.f, 0.f, 0.f, 0.f, 0.f};
  const float* arow = &ao[(long)(m0 + mrow) * INNER];
#pragma unroll 4
  for (int kk = 0; kk < INNER; kk += 4) {
    float2 av = *(const float2*)&arow[kk + koff];
    v2f a; a.x = av.x; a.y = av.y;
    v2f bb;
    bb.x = Wo[(long)(kk + koff) * DIM + n0 + mrow];
    bb.y = Wo[(long)(kk + koff + 1) * DIM + n0 + mrow];
    acc = wmma4(a, bb, acc);
  }
  float bias = bo[n0 + mrow];
#pragma unroll
  for (int r = 0; r < 8; ++r) {
    int mr = r + ((lane < 16) ? 0 : 8);
    out[(long)(m0 + mr) * DIM + n0 + mrow] = acc[r] + bias;
  }
}

// ---------------------------------------------------------------------------
extern "C" void kernel_launch(void* const* d_in, const int* in_sizes, int n_in,
                              void* d_out, int out_size, void* d_ws,
                              size_t ws_size, hipStream_t stream) {
  const float* nodes = (const float*)d_in[0];
  const float* edges = (const float*)d_in[1];
  const float* Wq = (const float*)d_in[2];
  const float* bq = (const float*)d_in[3];
  const float* Wkv = (const float*)d_in[4];
  const float* bkv = (const float*)d_in[5];
  const float* We = (const float*)d_in[6];
  const float* be = (const float*)d_in[7];
  const float* Wo = (const float*)d_in[8];
  const float* bo = (const float*)d_in[9];
  float* out = (float*)d_out;

  float* qkv = (float*)d_ws;                         // 768*1536
  float* wbuf = qkv + (long)ROWS * 1536;             // 6144*128
  float* qbe = wbuf + (long)B_SZ * HEADS * N_SEQ * EDIM;  // 6144
  float* ao = qbe + (long)B_SZ * HEADS * N_SEQ;      // 768*512

  proj_qkv_kernel<<<576, 256, 0, stream>>>(nodes, Wq, bq, Wkv, bkv, qkv);
  rope_kernel<<<ROWS, 256, 0, stream>>>(qkv);
  proj_w_kernel<<<384, 256, 0, stream>>>(qkv, We, wbuf);
  qbe_kernel<<<24, 256, 0, stream>>>(qkv, be, qbe);
  attn_kernel<<<dim3(N_SEQ / TI, HEADS, B_SZ), 256, 0, stream>>>(
      edges, qkv, wbuf, qbe, We, be, ao);
  out_proj_kernel<<<96, 256, 0, stream>>>(ao, Wo, bo, out);
}